// MaskAugmentation_58308476010519
// MI455X (gfx1250) — compile-verified
//
#include <hip/hip_runtime.h>
#include <hip/hip_bf16.h>
#include <stdint.h>

// ---------------------------------------------------------------------------
// MaskAugmentation for MI455X (gfx1250, wave32).
//  - One wave32 per row (L = 200 elements, <= 8 slots of 32 lanes).
//  - Row staged Global->LDS with the CDNA5 Tensor Data Mover (TENSORcnt),
//    falling back to direct global loads if the builtin is unavailable.
//  - Exact k-of-n uniform selection via repeated wave-min reduction on
//    hashed scores (position tiebreak in low 9 bits => unique argmin).
// Memory-bound: ~105 MB total traffic => ~4.5us at 23.3 TB/s.
// ---------------------------------------------------------------------------

#define WAVES_PER_BLOCK 8
#define ROW_STRIDE      256   // ints of LDS per row (L <= 256)
#define MAX_SLOTS       8     // ceil(256/32)

#if defined(__has_builtin)
#  if __has_builtin(__builtin_amdgcn_tensor_load_to_lds) && \
      __has_builtin(__builtin_amdgcn_s_wait_tensorcnt)
#    define HAVE_TDM 1
#  endif
#endif
#ifndef HAVE_TDM
#  define HAVE_TDM 0
#endif

typedef __attribute__((ext_vector_type(4))) unsigned int u32x4;
typedef __attribute__((ext_vector_type(8))) int          i32x8;
typedef __attribute__((ext_vector_type(4))) int          i32x4;

__device__ __forceinline__ unsigned pcg_hash(unsigned x) {
  x = x * 747796405u + 2891336453u;
  x = ((x >> ((x >> 28u) + 4u)) ^ x) * 277803737u;
  return (x >> 22u) ^ x;
}

__global__ __launch_bounds__(256) void MaskAugmentation_58308476010519_kernel(
    const int* __restrict__ seq, const int* __restrict__ slen,
    float* __restrict__ out, float* __restrict__ outLen, int B, int L) {
  __shared__ int rows[WAVES_PER_BLOCK * ROW_STRIDE];

  const int lane = (int)(threadIdx.x & 31u);
  const int wave = (int)(threadIdx.x >> 5u);
  const int r    = (int)blockIdx.x * WAVES_PER_BLOCK + wave;
  if (r >= B) return;  // wave-uniform exit

  const size_t rowBase = (size_t)r * (size_t)L;
  const int    sl      = slen[r];  // wave-uniform scalar load

#if HAVE_TDM
  {
    // This kernel's only LDS object -> allocated at LDS offset 0, so the
    // per-wave row buffer sits at a statically known byte offset.
    const unsigned lds_off = (unsigned)(wave * ROW_STRIDE * (int)sizeof(int));
    const uint64_t gaddr   = (uint64_t)(const void*)(seq + rowBase);

    // D# group 0: count=1 | lds_addr | global_addr[56:0] | type=2 ("image")
    u32x4 g0 = { 1u,
                 lds_off,
                 (unsigned)(gaddr & 0xFFFFFFFFu),
                 (unsigned)((gaddr >> 32) & 0x01FFFFFFu) | (2u << 30) };
    // D# group 1: data_size=4B, tensor_dim0=L, tensor_dim1=1,
    //             tile_dim0=L, tile_dim1=1, tensor_dim0_stride=L
    i32x8 g1 = { (int)(2u << 16),                         // data_size = 2 (4B)
                 (int)(((unsigned)L & 0xFFFFu) << 16),    // tensor_dim0 lo16
                 (int)(1u << 16),                         // dim0 hi=0 | tensor_dim1=1
                 (int)(((unsigned)L & 0xFFFFu) << 16),    // dim1 hi=0 | tile_dim0=L
                 1,                                       // tile_dim1=1, tile_dim2=0
                 L,                                       // tensor_dim0_stride lo32
                 0, 0 };
    i32x4 g2 = { 0, 0, 0, 0 };          // D# group 2 (unused, 2D tensor)
    i32x4 g3 = { 0, 0, 0, 0 };          // D# group 3 (unused, 2D tensor)
    i32x8 g4 = { 0, 0, 0, 0, 0, 0, 0, 0 };  // extra group (clang-23 6-arg form)
    __builtin_amdgcn_tensor_load_to_lds(g0, g1, g2, g3, g4, 0);
    __builtin_amdgcn_s_wait_tensorcnt(0);  // TDM done -> LDS tile valid
  }
  const volatile int* myRow = &rows[wave * ROW_STRIDE];
#else
  (void)rows;
#endif

  // ---- load slots, build validity + hashed scores --------------------------
  int      vals[MAX_SLOTS];
  unsigned scores[MAX_SLOTS];
  int      myCnt = 0;
#pragma unroll
  for (int s = 0; s < MAX_SLOTS; ++s) {
    const int p = s * 32 + lane;
    int  v  = 0;
    bool ok = false;
    if (p < L) {
#if HAVE_TDM
      v = myRow[p];
#else
      v = seq[rowBase + (size_t)p];
#endif
      ok = (p < sl) && (v != 0);
    }
    vals[s] = v;
    const unsigned h = pcg_hash(((unsigned)r << 8) | (unsigned)p);
    scores[s] = ok ? ((h & 0xFFFFFE00u) | (unsigned)p) : 0xFFFFFFFFu;
    myCnt += ok ? 1 : 0;
  }

  // ---- n_valid via wave32 reduction ---------------------------------------
  int nv = myCnt;
#pragma unroll
  for (int off = 16; off > 0; off >>= 1) nv += __shfl_xor(nv, off, 32);

  // floor(n*0.2f) == n/5 for n in [0,200]
  int nm = nv / 5;
  nm = (nm < 1) ? 1 : nm;
  nm = (nm > nv) ? nv : nm;
  if (sl <= 1 || nv == 0) nm = 0;

  // ---- pick nm smallest scores: repeated wave-min (exact, unique keys) ----
  unsigned picked = 0u;
  for (int it = 0; it < nm; ++it) {
    unsigned m = 0xFFFFFFFFu;
#pragma unroll
    for (int s = 0; s < MAX_SLOTS; ++s) m = (scores[s] < m) ? scores[s] : m;
#pragma unroll
    for (int off = 16; off > 0; off >>= 1) {
      const unsigned o = (unsigned)__shfl_xor((int)m, off, 32);
      m = (o < m) ? o : m;
    }
    const unsigned pmin = m & 511u;          // position of the winner
    if (lane == (int)(pmin & 31u)) {
      const int os = (int)(pmin >> 5u);
      scores[os] = 0xFFFFFFFFu;              // retire
      picked |= (1u << os);
    }
  }

  // ---- write augmented row + seq_len (harness output dtype: float) -------
#pragma unroll
  for (int s = 0; s < MAX_SLOTS; ++s) {
    const int p = s * 32 + lane;
    if (p < L) {
      const bool msk = (picked >> s) & 1u;
      out[rowBase + (size_t)p] = msk ? 0.0f : (float)vals[s];
    }
  }
  if (lane == 0) outLen[r] = (float)sl;
}

extern "C" void kernel_launch(void* const* d_in, const int* in_sizes, int n_in,
                              void* d_out, int out_size, void* d_ws, size_t ws_size,
                              hipStream_t stream) {
  (void)n_in; (void)d_ws; (void)ws_size; (void)out_size;
  const int* seq  = (const int*)d_in[0];
  const int* slen = (const int*)d_in[1];
  const int  B    = in_sizes[1];
  const int  L    = in_sizes[0] / in_sizes[1];

  float* out    = (float*)d_out;
  float* outLen = out + (size_t)B * (size_t)L;

  const int blocks = (B + WAVES_PER_BLOCK - 1) / WAVES_PER_BLOCK;
  MaskAugmentation_58308476010519_kernel<<<blocks, 256, 0, stream>>>(
      seq, slen, out, outLen, B, L);
}